// DNC_read_head_33500744908901
// MI455X (gfx1250) — compile-verified
//
#include <hip/hip_runtime.h>
#include <math.h>

typedef float v2f __attribute__((ext_vector_type(2)));
typedef float v8f __attribute__((ext_vector_type(8)));
typedef int   v4i_ls __attribute__((__vector_size__(16)));   // matches builtin param

constexpr int B_ = 32, N_ = 1024, M_ = 128, H_ = 2;
constexpr int BN = B_ * N_;
constexpr float EPS = 1e-8f;

#if defined(__HIP_DEVICE_COMPILE__) && defined(__has_builtin)
#  if __has_builtin(__builtin_amdgcn_global_load_async_to_lds_b128) && \
      __has_builtin(__builtin_amdgcn_s_wait_asynccnt)
#    define USE_ASYNC_LDS 1
#  endif
#endif
#ifndef USE_ASYNC_LDS
#  define USE_ASYNC_LDS 0
#endif

// ---------------- init: zero the b-vector accumulator ----------------
__global__ void k_init(float* __restrict__ bv) {
  int i = blockIdx.x * 256 + threadIdx.x;
  if (i < BN) bv[i] = 0.f;
}

// ---------------- scores: s[b,n] = beta * (mem_row . k) / (|mem_row||k| + eps)
__global__ void k_scores(const float* __restrict__ read_keys,
                         const float* __restrict__ read_strengths,
                         const float* __restrict__ memory,
                         const int* __restrict__ head_ptr,
                         float* __restrict__ scores) {
  const int head = *head_ptr;
  const int lane = threadIdx.x & 31;
  const int wv   = threadIdx.x >> 5;
  const int g    = blockIdx.x * 8 + wv;     // [0, B*N)
  const int b    = g >> 10;
  const int n    = g & 1023;

  const float4* mrow = (const float4*)(memory + (size_t)(b * N_ + n) * M_);
  const float4* krow = (const float4*)(read_keys + (size_t)(b * H_ + head) * M_);
  float4 m4 = mrow[lane];
  float4 k4 = krow[lane];

  float dot = m4.x * k4.x + m4.y * k4.y + m4.z * k4.z + m4.w * k4.w;
  float msq = m4.x * m4.x + m4.y * m4.y + m4.z * m4.z + m4.w * m4.w;
  float ksq = k4.x * k4.x + k4.y * k4.y + k4.z * k4.z + k4.w * k4.w;
  #pragma unroll
  for (int off = 16; off > 0; off >>= 1) {
    dot += __shfl_xor(dot, off, 32);
    msq += __shfl_xor(msq, off, 32);
    ksq += __shfl_xor(ksq, off, 32);
  }
  float beta = read_strengths[b * H_ + head];
  float s = beta * dot / (sqrtf(msq) * sqrtf(ksq) + EPS);
  if (lane == 0) scores[g] = s;
}

// ---------------- softmax over N per batch (in place) ----------------
__global__ void k_softmax(float* __restrict__ sc) {
  __shared__ float red[256];
  const int b = blockIdx.x, t = threadIdx.x;
  float v[4];
  float mx = -INFINITY;
  #pragma unroll
  for (int i = 0; i < 4; ++i) { v[i] = sc[b * N_ + t + 256 * i]; mx = fmaxf(mx, v[i]); }
  red[t] = mx; __syncthreads();
  for (int s = 128; s > 0; s >>= 1) { if (t < s) red[t] = fmaxf(red[t], red[t + s]); __syncthreads(); }
  mx = red[0]; __syncthreads();
  float e[4], sum = 0.f;
  #pragma unroll
  for (int i = 0; i < 4; ++i) { e[i] = __expf(v[i] - mx); sum += e[i]; }
  red[t] = sum; __syncthreads();
  for (int s = 128; s > 0; s >>= 1) { if (t < s) red[t] += red[t + s]; __syncthreads(); }
  float inv = 1.f / red[0];
  #pragma unroll
  for (int i = 0; i < 4; ++i) sc[b * N_ + t + 256 * i] = e[i] * inv;
}

// ---------------- WMMA core on one staged 16x32 slab (2 j-tiles) ----------------
constexpr int STRIDE = 44;              // padded LDS row stride (floats), 16x32 slab
constexpr int SLABF  = 16 * STRIDE;     // floats per slab buffer

__device__ __forceinline__ void wmma_slab32(const float* sl,          // LDS slab
                                            const float* wsh,        // LDS w
                                            float* bacc,             // LDS b accum
                                            const v2f aw[4], v8f& cf,
                                            int j0, int lane, int half, int l16) {
  #pragma unroll
  for (int jt = 0; jt < 2; ++jt) {
    const int jc = jt * 16;
    v8f cb = {};
    #pragma unroll
    for (int c = 0; c < 4; ++c) {
      // f-direction: A = tile column-chunk (16x4), B = w-chunk broadcast into all cols
      const float* ap = sl + l16 * STRIDE + jc + 4 * c + 2 * half;
      v2f a;  a[0] = ap[0];  a[1] = ap[1];
      v2f bw;
      bw[0] = wsh[j0 + jc + 4 * c + 2 * half + 0];
      bw[1] = wsh[j0 + jc + 4 * c + 2 * half + 1];
      cf = __builtin_amdgcn_wmma_f32_16x16x4_f32(false, a, false, bw,
                                                 (short)0, cf, false, false);
      // b-direction: A = w-chunk broadcast into all rows, B = tile row-chunk (4x16)
      v2f bt;
      bt[0] = sl[(4 * c + 2 * half + 0) * STRIDE + jc + l16];
      bt[1] = sl[(4 * c + 2 * half + 1) * STRIDE + jc + l16];
      cb = __builtin_amdgcn_wmma_f32_16x16x4_f32(false, aw[c], false, bt,
                                                 (short)0, cb, false, false);
    }
    if (lane < 16) atomicAdd(&bacc[j0 + jc + lane], cb[0]);
  }
}

// ---------------- L pass: f = L.w and b = L^T.w in ONE read of L ----------------
__global__ __launch_bounds__(256)
void k_Lpass(const float* __restrict__ L,
             const float* __restrict__ W_old,
             float* __restrict__ f_out,
             float* __restrict__ b_out) {
  __shared__ float wsh[N_];                  // W_old[b][:]
  __shared__ float bacc[N_];                 // b-direction accumulator
  __shared__ float slab[8][2][SLABF];        // per-wave double-buffered 16x32 stage

  const int b    = blockIdx.x >> 3;
  const int iblk = blockIdx.x & 7;
  const int lane = threadIdx.x & 31;
  const int wv   = threadIdx.x >> 5;
  const int i0   = iblk * 128 + wv * 16;     // this wave's 16 rows of L[b]
  const int half = lane >> 4;                // 0: lanes 0-15, 1: lanes 16-31
  const int l16  = lane & 15;

  const float* Lb = L + (size_t)b * N_ * N_;
  const float* w  = W_old + b * N_;

  for (int i = threadIdx.x; i < N_; i += 256) { wsh[i] = w[i]; bacc[i] = 0.f; }
  __syncthreads();

  // b-direction A operand: w[i0..i0+15] broadcast across rows, constant over j loop.
  v2f aw[4];
  #pragma unroll
  for (int c = 0; c < 4; ++c) {
    aw[c][0] = wsh[i0 + 4 * c + 2 * half + 0];
    aw[c][1] = wsh[i0 + 4 * c + 2 * half + 1];
  }

  v8f cf = {};  // f accumulator persists across the full j loop (no atomics for f)

  // staging geometry: 16 rows x 32 cols; 4 b128 transfers/slab
  const int srow = lane >> 3;        // 0..3 (row within group of 4)
  const int scol = (lane & 7) * 4;   // float column 0..28

#if USE_ASYNC_LDS
  // ---- async global->LDS, double-buffered, constant buffer indices so the
  //      compute path keeps AS(3) inference (ds_load, not flat_load) ----
  auto issue = [&](int bufi, int j0) {
    #pragma unroll
    for (int t = 0; t < 4; ++t) {
      const int row = 4 * t + srow;
      __builtin_amdgcn_global_load_async_to_lds_b128(
          (__attribute__((address_space(1))) v4i_ls*)
              (Lb + (size_t)(i0 + row) * N_ + j0 + scol),
          (__attribute__((address_space(3))) v4i_ls*)
              (&slab[wv][bufi][row * STRIDE + scol]),
          0, 0);
    }
  };
  issue(0, 0);
  issue(1, 32);
  for (int k = 0; k < N_ / 64; ++k) {
    const int j0 = k * 64;
    if (j0 + 128 < N_)
      __builtin_prefetch(Lb + (size_t)(i0 + srow) * N_ + (j0 + 128) + scol, 0, 1);

    __builtin_amdgcn_s_wait_asynccnt(4);            // buf0 (oldest 4) landed
    wmma_slab32(slab[wv][0], wsh, bacc, aw, cf, j0, lane, half, l16);
    if (k + 1 < N_ / 64) {
      issue(0, j0 + 64);                            // refill buf0
      __builtin_amdgcn_s_wait_asynccnt(4);          // buf1 (now oldest) landed
    } else {
      __builtin_amdgcn_s_wait_asynccnt(0);          // drain: buf1 landed
    }
    wmma_slab32(slab[wv][1], wsh, bacc, aw, cf, j0 + 32, lane, half, l16);
    if (k + 1 < N_ / 64)
      issue(1, j0 + 96);                            // refill buf1
  }
#else
  // ---- fallback: VGPR staging with per-iteration barrier ----
  for (int j0 = 0; j0 < N_; j0 += 32) {
    #pragma unroll
    for (int t = 0; t < 4; ++t) {
      const int row = 4 * t + srow;
      const float4 v4 = *(const float4*)(Lb + (size_t)(i0 + row) * N_ + j0 + scol);
      *(float4*)(&slab[wv][0][row * STRIDE + scol]) = v4;
    }
    if (j0 + 32 < N_)
      __builtin_prefetch(Lb + (size_t)(i0 + srow) * N_ + (j0 + 32) + scol, 0, 1);
    __syncthreads();
    wmma_slab32(slab[wv][0], wsh, bacc, aw, cf, j0, lane, half, l16);
    __syncthreads();
  }
#endif

  // f extraction: column 0 lives in lane 0 (M=0..7) and lane 16 (M=8..15)
  if (l16 == 0) {
    float* fp = f_out + b * N_ + i0 + half * 8;
    fp[0] = cf[0]; fp[1] = cf[1]; fp[2] = cf[2]; fp[3] = cf[3];
    fp[4] = cf[4]; fp[5] = cf[5]; fp[6] = cf[6]; fp[7] = cf[7];
  }
  __syncthreads();
  for (int i = threadIdx.x; i < N_; i += 256)
    atomicAdd(&b_out[b * N_ + i], bacc[i]);
}

// ---------------- blend: W = rm0*b + rm1*C + rm2*f ----------------
__global__ void k_combine(const float* __restrict__ read_mode,
                          const int* __restrict__ head_ptr,
                          const float* __restrict__ C,
                          const float* __restrict__ f,
                          const float* __restrict__ bv,
                          float* __restrict__ out) {
  int idx = blockIdx.x * 256 + threadIdx.x;
  if (idx >= BN) return;
  int b = idx >> 10;
  int head = *head_ptr;
  const float* rm = read_mode + (b * H_ + head) * 3;
  out[idx] = rm[0] * bv[idx] + rm[1] * C[idx] + rm[2] * f[idx];
}

// ---------------- mem_content[b,m] = sum_n W[b,n] * memory[b,n,m] ----------------
__global__ void k_content(const float* __restrict__ memory,
                          const float* __restrict__ W,
                          float* __restrict__ out2) {
  const int b = blockIdx.x, m = threadIdx.x;
  const float* mb = memory + (size_t)b * N_ * M_;
  const float* wb = W + b * N_;
  float acc = 0.f;
  for (int n = 0; n < N_; ++n)
    acc = fmaf(wb[n], mb[(size_t)n * M_ + m], acc);
  out2[b * M_ + m] = acc;
}

extern "C" void kernel_launch(void* const* d_in, const int* in_sizes, int n_in,
                              void* d_out, int out_size, void* d_ws, size_t ws_size,
                              hipStream_t stream) {
  (void)in_sizes; (void)n_in; (void)out_size; (void)ws_size;
  const float* read_keys      = (const float*)d_in[0];
  const float* read_strengths = (const float*)d_in[1];
  const float* read_mode      = (const float*)d_in[2];
  const float* W_old          = (const float*)d_in[3];
  const float* L              = (const float*)d_in[4];
  const float* memory         = (const float*)d_in[5];
  const int*   head_ptr       = (const int*)d_in[6];

  float* ws = (float*)d_ws;
  float* sc = ws;            // scores, softmaxed in place -> C
  float* f  = ws + BN;       // f = L.w
  float* bv = ws + 2 * BN;   // b = L^T.w (atomically accumulated)
  float* outW = (float*)d_out;

  k_init   <<<BN / 256, 256, 0, stream>>>(bv);
  k_scores <<<BN / 8,   256, 0, stream>>>(read_keys, read_strengths, memory, head_ptr, sc);
  k_softmax<<<B_,       256, 0, stream>>>(sc);
  k_Lpass  <<<B_ * 8,   256, 0, stream>>>(L, W_old, f, bv);
  k_combine<<<BN / 256, 256, 0, stream>>>(read_mode, head_ptr, sc, f, bv, outW);
  k_content<<<B_,       128, 0, stream>>>(memory, outW, outW + BN);
}